// VariationalQuantumCircuit_23579370455598
// MI455X (gfx1250) — compile-verified
//
#include <hip/hip_runtime.h>
#include <math.h>

#define D        4096
#define BATCH    16384
#define NLAYERS  3
#define NGROUPS  6
#define TPB      256
#define PER_T    (D / TPB)       // 16 scalar elements per thread
#define NV4      (D / 4 / TPB)   // 4 float4s per thread
#define NWAVES   (TPB / 32)      // 8 wave32s
#define EPSF     1e-8f

typedef float v4f __attribute__((ext_vector_type(4)));

// Deterministic block-wide sum: wave32 shuffle tree, then fixed-order
// accumulation of the 8 wave partials by thread 0. No atomics.
__device__ __forceinline__ float block_reduce_sum(float v, volatile float* red) {
#pragma unroll
    for (int off = 16; off > 0; off >>= 1)
        v += __shfl_xor(v, off, 32);
    const int lane = threadIdx.x & 31;
    const int wid  = threadIdx.x >> 5;
    if (lane == 0) red[wid] = v;
    __syncthreads();
    if (threadIdx.x == 0) {
        float tot = 0.0f;
#pragma unroll
        for (int i = 0; i < NWAVES; ++i) tot += red[i];
        red[NWAVES] = tot;
    }
    __syncthreads();
    float r = red[NWAVES];
    __syncthreads();   // allow red[] to be reused by the next reduction
    return r;
}

__global__ __launch_bounds__(TPB) void vqc_kernel(
    const float* __restrict__ x,     // BATCH x D
    const float* __restrict__ qp,    // >= 18 params used
    const float* __restrict__ w,     // D x 1
    const float* __restrict__ b,     // 1
    float* __restrict__ meas,        // BATCH x D
    float* __restrict__ cls)         // BATCH
{
    __shared__ __align__(16) float bufA[D];
    __shared__ __align__(16) float bufB[D];
    __shared__ __align__(16) float wbuf[D];
    __shared__ float cs[NLAYERS * NGROUPS];
    __shared__ float sn[NLAYERS * NGROUPS];
    __shared__ float red[NWAVES + 1];

    const int t   = threadIdx.x;
    const int row = blockIdx.x;
    const float* xr = x + (size_t)row * D;

    // Precompute the 18 rotation angles once per block.
    if (t < NLAYERS * NGROUPS) {
        float a = qp[t];
        cs[t] = cosf(a);
        sn[t] = sinf(a);
    }

    // ---- CDNA5 async global->LDS staging (ASYNCcnt path) ----
    // Row of x (16 KB, read-once -> non-temporal) and w_out (16 KB, shared
    // across all blocks -> default caching). GVS mode: saddr = base (SGPR
    // pair), vaddr = per-lane byte offset, vdst = per-lane LDS byte address.
    {
        unsigned laX = (unsigned)(size_t)(&bufA[0]);
        unsigned laW = (unsigned)(size_t)(&wbuf[0]);
        unsigned long long gX = (unsigned long long)(size_t)xr;
        unsigned long long gW = (unsigned long long)(size_t)w;
#pragma unroll
        for (int k = 0; k < NV4; ++k) {
            unsigned off   = (unsigned)((k * TPB + t) * 16);
            unsigned laddr = laX + off;
            asm volatile("global_load_async_to_lds_b128 %0, %1, %2 th:TH_LOAD_NT"
                         :
                         : "v"(laddr), "v"(off), "s"(gX)
                         : "memory");
        }
#pragma unroll
        for (int k = 0; k < NV4; ++k) {
            unsigned off   = (unsigned)((k * TPB + t) * 16);
            unsigned laddr = laW + off;
            asm volatile("global_load_async_to_lds_b128 %0, %1, %2"
                         :
                         : "v"(laddr), "v"(off), "s"(gW)
                         : "memory");
        }
        asm volatile("s_wait_asynccnt 0x0" ::: "memory");
    }
    __syncthreads();

    const v4f* bufA4 = (const v4f*)bufA;
    const v4f* bufB4 = (const v4f*)bufB;
    const v4f* wbuf4 = (const v4f*)wbuf;
    v4f* outA4 = (v4f*)bufA;

    // ---- initial row norm: qs = x / (||x|| + eps) (folded into layer 1) ----
    float ss = 0.0f;
#pragma unroll
    for (int k = 0; k < NV4; ++k) {
        v4f v = bufA4[k * TPB + t];
        v4f p = v * v;
        ss += p.x + p.y + p.z + p.w;
    }
    ss = block_reduce_sum(ss, red);
    float scl = 1.0f / (sqrtf(ss) + EPSF);

    // ---- 3 layers: segment rotations (A->B) then circulant stencil (B->A) --
    for (int layer = 0; layer < NLAYERS; ++layer) {
        const float* csl = &cs[layer * NGROUPS];
        const float* snl = &sn[layer * NGROUPS];

        // Rotation pass (scalar): element j is in a segment iff its top set
        // bit position p is even and p <= 10. Segment = [2^p, 2^(p+1)).
#pragma unroll
        for (int k = 0; k < PER_T; ++k) {
            int j = t + k * TPB;
            float vj  = bufA[j] * scl;
            float out = vj;
            if (j > 0) {
                int p = 31 - __clz(j);
                if (((p & 1) == 0) && (p < 11)) {
                    int len  = 1 << p;
                    int lo   = len;
                    int prev = lo + ((j - lo - 1) & (len - 1));
                    int g    = p >> 1;
                    out = vj * csl[g] + (bufA[prev] * scl) * snl[g];
                }
            }
            bufB[j] = out;
        }
        __syncthreads();

        // qs @ (I + 0.1*(circ shifts)) == 3-point circular stencil, vectorized.
#pragma unroll
        for (int k = 0; k < NV4; ++k) {
            int vb = k * TPB + t;
            int e  = vb * 4;
            v4f c = bufB4[vb];
            float left  = bufB[(e + D - 1) & (D - 1)];
            float right = bufB[(e + 4) & (D - 1)];
            v4f o;
            o.x = c.x + 0.1f * (left + c.y);
            o.y = c.y + 0.1f * (c.x + c.z);
            o.z = c.z + 0.1f * (c.y + c.w);
            o.w = c.w + 0.1f * (c.z + right);
            outA4[vb] = o;
        }
        __syncthreads();
        scl = 1.0f;  // initial normalization only applies to layer 0 input
    }

    // ---- final renorm, square, write measurements, fused output matvec ----
    float ss2 = 0.0f;
#pragma unroll
    for (int k = 0; k < NV4; ++k) {
        v4f v = bufA4[k * TPB + t];
        v4f p = v * v;
        ss2 += p.x + p.y + p.z + p.w;
    }
    ss2 = block_reduce_sum(ss2, red);
    float s2 = 1.0f / (sqrtf(ss2) + EPSF);

    float dot = 0.0f;
    v4f* mrow4 = (v4f*)(meas + (size_t)row * D);
#pragma unroll
    for (int k = 0; k < NV4; ++k) {
        int vb = k * TPB + t;
        v4f q = bufA4[vb] * s2;
        v4f m = q * q;
        __builtin_nontemporal_store(m, &mrow4[vb]);  // 268 MB stream: keep L2 clean
        v4f pr = m * wbuf4[vb];                      // staged w, own-lane data
        dot += pr.x + pr.y + pr.z + pr.w;
    }
    dot = block_reduce_sum(dot, red);

    if (t == 0) {
        float logit = dot + b[0];
        cls[row] = 1.0f / (1.0f + expf(-logit));
    }
}

extern "C" void kernel_launch(void* const* d_in, const int* in_sizes, int n_in,
                              void* d_out, int out_size, void* d_ws, size_t ws_size,
                              hipStream_t stream) {
    const float* x  = (const float*)d_in[0];   // (16384, 4096) f32
    const float* qp = (const float*)d_in[1];   // (108,) f32, first 18 used
    const float* w  = (const float*)d_in[2];   // (4096, 1) f32
    const float* b  = (const float*)d_in[3];   // (1,) f32

    float* meas = (float*)d_out;                         // 16384*4096
    float* cls  = (float*)d_out + (size_t)BATCH * D;     // 16384

    vqc_kernel<<<dim3(BATCH), dim3(TPB), 0, stream>>>(x, qp, w, b, meas, cls);
}